// LieAttention_41154376630712
// MI455X (gfx1250) — compile-verified
//
#include <hip/hip_runtime.h>
#include <hip/hip_bf16.h>

typedef __attribute__((ext_vector_type(2))) float v2f;
typedef __attribute__((ext_vector_type(8))) float v8f;

#define Bq 4
#define Lq 1024
#define Sq 1024
#define Hq 8
#define Eq 64

// ---------------------------------------------------------------------------
// Kernel 1: fused  comm = Q·K^T - K·Q^T  -> tanh -> gelu -> gumbel-sigmoid
//           -> masks -> A (written to d_out) + per-row entropy.
// One wave (32 lanes) owns one 16-row l-tile and sweeps all 64 s-tiles.
// WMMA f32 16x16x4; both products produced in C-layout so the antisymmetric
// difference is a plain vector subtract.
// ---------------------------------------------------------------------------
__global__ __launch_bounds__(128)
void lie_attn_scores_kernel(const float* __restrict__ query,
                            const float* __restrict__ key,
                            const float* __restrict__ phi,
                            const float* __restrict__ log_gain,
                            const float* __restrict__ log_tau,
                            const float* __restrict__ log_tau_comm,
                            const float* __restrict__ hard_mask,
                            const float* __restrict__ u,
                            float* __restrict__ A_out,
                            float* __restrict__ ent_out) {
    const int lane = threadIdx.x & 31;
    const int wv   = threadIdx.x >> 5;          // 4 waves / block
    const int half = lane >> 4;                 // K-pair select for A/B frags
    const int lr   = lane & 15;                 // M (A-frag) / N (B,C frags)

    int bid  = blockIdx.x;                      // B*H*16 blocks
    int lblk = bid & 15;
    int bh   = bid >> 4;
    int h    = bh & (Hq - 1);
    int b    = bh / Hq;
    int l_base = (lblk * 4 + wv) * 16;

    // scalar parameters (torch clamps)
    float gain     = fminf(fmaxf(__expf(log_gain[0]),     1e-3f), 1000.0f);
    float tau_comm = fminf(fmaxf(__expf(log_tau_comm[0]), 1e-3f), 10.0f);
    float tau      = fminf(fmaxf(__expf(log_tau[0]),      0.1f),  5.0f);
    float coef     = gain / tau_comm;
    float inv_tau  = 1.0f / tau;

    // Preload A-operand fragments: Q rows (for acc1) and K rows (for acc2)
    // of this l-tile.  A-frag 16x4: lane=M, half selects K-pair.
    const float* qrow_l = query + (((size_t)b * Lq + l_base + lr) * Hq + h) * Eq;
    const float* krow_l = key   + (((size_t)b * Sq + l_base + lr) * Hq + h) * Eq;
    v2f qa[16], ka[16];
#pragma unroll
    for (int kk = 0; kk < 16; ++kk) {
        qa[kk] = *(const v2f*)(qrow_l + 4 * kk + 2 * half);
        ka[kk] = *(const v2f*)(krow_l + 4 * kk + 2 * half);
    }

    float ent[8];
#pragma unroll
    for (int v = 0; v < 8; ++v) ent[v] = 0.0f;

    const size_t hls_base = ((size_t)h * Lq) * Sq;          // phi/u slice
    const size_t A_bh     = ((size_t)(b * Hq + h) * Lq) * Sq;

    for (int st = 0; st < Sq / 16; ++st) {
        int s_base = st * 16;
        const float* krow_s = key   + (((size_t)b * Sq + s_base + lr) * Hq + h) * Eq;
        const float* qrow_s = query + (((size_t)b * Lq + s_base + lr) * Hq + h) * Eq;

        v8f acc1 = {0.f,0.f,0.f,0.f,0.f,0.f,0.f,0.f};
        v8f acc2 = {0.f,0.f,0.f,0.f,0.f,0.f,0.f,0.f};
#pragma unroll
        for (int kk = 0; kk < 16; ++kk) {
            v2f b1 = *(const v2f*)(krow_s + 4 * kk + 2 * half);   // K_s  (B-frag)
            v2f b2 = *(const v2f*)(qrow_s + 4 * kk + 2 * half);   // Q_s  (B-frag)
            acc1 = __builtin_amdgcn_wmma_f32_16x16x4_f32(
                       false, qa[kk], false, b1, (short)0, acc1, false, false);
            acc2 = __builtin_amdgcn_wmma_f32_16x16x4_f32(
                       false, ka[kk], false, b2, (short)0, acc2, false, false);
        }
        v8f comm = acc1 - acc2;                 // antisymmetric term, C-layout

#pragma unroll
        for (int v = 0; v < 8; ++v) {
            int m = l_base + v + 8 * half;      // global l row
            int n = s_base + lr;                // global s col
            float x = tanhf(coef * comm[v]);
            float g = 0.5f * x * (1.0f + erff(x * 0.70710678118654752f)); // exact gelu
            float att = 0.125f * g;             // scale = 1/sqrt(64)

            size_t hls = hls_base + (size_t)m * Sq + n;
            float uu = u[hls];
            float ph = phi[hls];
            float logit = (__logf(uu + 1e-8f) - __logf(1.0f - uu + 1e-8f) + ph) * inv_tau;
            float mrel  = 1.0f / (1.0f + __expf(-logit));
            att *= mrel * hard_mask[(size_t)m * Sq + n];

            A_out[A_bh + (size_t)m * Sq + n] = att;
            ent[v] -= att * __logf(fmaxf(att, 1e-8f));
        }
    }

    // entropy: reduce each row across the 16 lanes of its half-wave
#pragma unroll
    for (int v = 0; v < 8; ++v) {
        float e = ent[v];
        e += __shfl_xor(e, 1);
        e += __shfl_xor(e, 2);
        e += __shfl_xor(e, 4);
        e += __shfl_xor(e, 8);
        if (lr == 0) {
            int m = l_base + v + 8 * half;
            ent_out[(size_t)(b * Hq + h) * Lq + m] = e;
        }
    }
}

// ---------------------------------------------------------------------------
// Kernel 2: V[b,l,h,d] = sum_s A[b,h,l,s] * value[b,s,h,d]
// One wave per 16-row l-tile, computing the full 16x64 output strip
// (4 WMMA accumulators).  K-loop over S=1024 in steps of 4.
// ---------------------------------------------------------------------------
__global__ __launch_bounds__(128)
void lie_attn_av_kernel(const float* __restrict__ A_in,
                        const float* __restrict__ value,
                        float* __restrict__ V_out) {
    const int lane = threadIdx.x & 31;
    const int wv   = threadIdx.x >> 5;
    const int half = lane >> 4;
    const int lr   = lane & 15;

    int bid  = blockIdx.x;
    int lblk = bid & 15;
    int bh   = bid >> 4;
    int h    = bh & (Hq - 1);
    int b    = bh / Hq;
    int l_base = (lblk * 4 + wv) * 16;

    const float* arow = A_in + (((size_t)(b * Hq + h) * Lq + l_base + lr) * Sq);
    const float* vbase = value + (((size_t)b * Sq) * Hq + h) * Eq;

    v8f acc[4];
#pragma unroll
    for (int dt = 0; dt < 4; ++dt)
        acc[dt] = (v8f){0.f,0.f,0.f,0.f,0.f,0.f,0.f,0.f};

#pragma unroll 4
    for (int k0 = 0; k0 < Sq; k0 += 4) {
        // A-frag: lane = l-row, half selects k-pair (contiguous b64 load)
        v2f a = *(const v2f*)(arow + k0 + 2 * half);
        __builtin_prefetch(arow + k0 + 128, 0, 0);   // global_prefetch_b8 ahead

        int k_lo = k0 + 2 * half;
        const float* vr0 = vbase + (size_t)k_lo * (Hq * Eq);
        const float* vr1 = vr0 + (Hq * Eq);
#pragma unroll
        for (int dt = 0; dt < 4; ++dt) {
            v2f bf;
            bf.x = vr0[dt * 16 + lr];
            bf.y = vr1[dt * 16 + lr];
            acc[dt] = __builtin_amdgcn_wmma_f32_16x16x4_f32(
                          false, a, false, bf, (short)0, acc[dt], false, false);
        }
    }

#pragma unroll
    for (int dt = 0; dt < 4; ++dt) {
#pragma unroll
        for (int v = 0; v < 8; ++v) {
            int m = l_base + v + 8 * half;
            int d = dt * 16 + lr;
            V_out[(((size_t)b * Lq + m) * Hq + h) * Eq + d] = acc[dt][v];
        }
    }
}

extern "C" void kernel_launch(void* const* d_in, const int* in_sizes, int n_in,
                              void* d_out, int out_size, void* d_ws, size_t ws_size,
                              hipStream_t stream) {
    (void)in_sizes; (void)n_in; (void)d_ws; (void)ws_size; (void)out_size;

    const float* query        = (const float*)d_in[0];
    const float* key          = (const float*)d_in[1];
    const float* value        = (const float*)d_in[2];
    const float* phi          = (const float*)d_in[3];
    const float* log_gain     = (const float*)d_in[4];
    const float* log_tau      = (const float*)d_in[5];
    const float* log_tau_comm = (const float*)d_in[6];
    const float* hard_mask    = (const float*)d_in[7];
    const float* u            = (const float*)d_in[8];
    // d_in[9..12] unused by the reference forward

    float* V_out   = (float*)d_out;                               // B*L*H*E
    float* A_out   = V_out + (size_t)Bq * Lq * Hq * Eq;           // B*H*L*S
    float* ent_out = A_out + (size_t)Bq * Hq * Lq * Sq;           // B*H*L

    dim3 grid(Bq * Hq * (Lq / 64));   // 512 blocks, 4 waves each -> 64 l-tiles/(b,h)
    dim3 block(128);

    lie_attn_scores_kernel<<<grid, block, 0, stream>>>(
        query, key, phi, log_gain, log_tau, log_tau_comm, hard_mask, u,
        A_out, ent_out);

    lie_attn_av_kernel<<<grid, block, 0, stream>>>(A_out, value, V_out);
}